// FullQueryLayer_20770461844117
// MI455X (gfx1250) — compile-verified
//
#include <hip/hip_runtime.h>
#include <math.h>

typedef __attribute__((ext_vector_type(2))) float v2f;
typedef __attribute__((ext_vector_type(8))) float v8f;
typedef __attribute__((ext_vector_type(4))) unsigned int u32x4;
typedef __attribute__((ext_vector_type(8))) int i32x8;
typedef __attribute__((ext_vector_type(4))) int i32x4;

#define B_  8
#define E_  128
#define Q_  128
#define S_  30720          // 96*320
#define STILE 128
#define LDSROW (STILE + 4) // pad 4 floats -> bank stride 4; conflict-free b64 frag loads
#define SCHUNK 512

#if defined(__has_builtin)
#if __has_builtin(__builtin_amdgcn_tensor_load_to_lds) && __has_builtin(__builtin_amdgcn_s_wait_tensorcnt)
#define HAVE_TDM 1
#endif
#endif

#ifdef HAVE_TDM
// 2D TDM load: rows x cols (4-byte elems), row stride in elems, LDS rows padded
// by 4 DWORDs every 128 DWORDs (pad_interval=6 -> 128 dw, pad_amount=3 -> 4 dw),
// i.e. LDS row stride = LDSROW floats. Descriptor per CDNA5 ISA 8.3/8.4.
// clang-23 lane: 6-arg builtin (g0, g1, g2, g3, g_extra, cpol).
__device__ __forceinline__ void tdm_load_2d(const float* gptr, unsigned lds_off,
                                            unsigned rows, unsigned cols,
                                            unsigned long long row_stride) {
    unsigned long long ga = (unsigned long long)(uintptr_t)gptr;
    u32x4 g0;
    g0[0] = 1u;                                        // count=1, user D#
    g0[1] = lds_off;                                   // lds_addr (bytes)
    g0[2] = (unsigned)(ga & 0xffffffffull);            // global_addr[31:0]
    g0[3] = ((unsigned)(ga >> 32) & 0x01ffffffu) | 0x80000000u; // addr[56:32] | type=2
    i32x8 g1;
    g1[0] = (2 << 16) | (1 << 20) | (6 << 22) | (3 << 25); // 4B elems, pad_en, 128dw/4dw
    g1[1] = (int)(cols << 16);                         // tensor_dim0[15:0]
    g1[2] = (int)(rows << 16);                         // tensor_dim0[31:16]=0 | tensor_dim1[15:0]
    g1[3] = (int)(cols << 16);                         // tensor_dim1[31:16]=0 | tile_dim0
    g1[4] = (int)rows;                                 // tile_dim1 | tile_dim2=0
    g1[5] = (int)(row_stride & 0xffffffffull);         // tensor_dim0_stride[31:0]
    g1[6] = (int)((row_stride >> 32) & 0xffffull);     // stride[47:32] | dim1_stride lo=0
    g1[7] = 0;
    i32x4 z4 = {};
    i32x8 z8 = {};
    __builtin_amdgcn_tensor_load_to_lds(g0, g1, z4, z4, z8, 0);
}
#endif

// ---------------------------------------------------------------------------
// Kernel 1: energy[b,q,s] = sum_e K[b,q,e] * x[b,e,s]
// K tile (Q_ x E_) -> LDS via TDM; x tile stored TRANSPOSED [s][e] so both A and
// B WMMA fragments are contiguous even-aligned ds_load_b64, conflict-free.
// ---------------------------------------------------------------------------
__global__ __launch_bounds__(256)
void gemm1_energy(const float* __restrict__ x, const float* __restrict__ Kq,
                  float* __restrict__ energy) {
    extern __shared__ float lds[];
    float* xt = lds;                     // [STILE][LDSROW]  x transposed: [s][e]
    float* kl = lds + STILE * LDSROW;    // [Q_][LDSROW]     K: [q][e]
    const int blk = blockIdx.x;
    const int b   = blk / (S_ / STILE);
    const int s0  = (blk % (S_ / STILE)) * STILE;
    const float* xb = x  + (size_t)b * E_ * S_;
    const float* Kb = Kq + (size_t)b * Q_ * E_;
    float*       eb = energy + (size_t)b * Q_ * S_;

    const int tid = threadIdx.x;
#ifdef HAVE_TDM
    if (tid < 32) {                      // wave 0 drives the Tensor Data Mover
        tdm_load_2d(Kb, (unsigned)(uintptr_t)kl, Q_, E_, E_);
    }
#else
    for (int idx = tid; idx < Q_ * (E_ / 4); idx += 256) {
        const int q  = idx >> 5;
        const int c4 = (idx & 31) << 2;
        float4 v = *(const float4*)(Kb + (size_t)q * E_ + c4);
        *(float4*)(&kl[q * LDSROW + c4]) = v;
    }
#endif
    // cooperative transpose-load of x tile: coalesced float4 read along s,
    // scattered b32 stores into [s][e]
    for (int idx = tid; idx < E_ * (STILE / 4); idx += 256) {
        const int e  = idx >> 5;
        const int c4 = (idx & 31) << 2;
        float4 v = *(const float4*)(xb + (size_t)e * S_ + s0 + c4);
        xt[(c4 + 0) * LDSROW + e] = v.x;
        xt[(c4 + 1) * LDSROW + e] = v.y;
        xt[(c4 + 2) * LDSROW + e] = v.z;
        xt[(c4 + 3) * LDSROW + e] = v.w;
    }
#ifdef HAVE_TDM
    if (tid < 32) __builtin_amdgcn_s_wait_tensorcnt(0);
#endif
    __syncthreads();

    const int wave = tid >> 5, lane = tid & 31;
    const int hi = lane >> 4, lo = lane & 15;
    const float* arow = &kl[(16 * wave + lo) * LDSROW];

    v8f acc[8] = {};
#pragma unroll 4
    for (int stp = 0; stp < E_ / 4; ++stp) {
        const int k0 = 4 * stp + 2 * hi;                 // e-pair for this lane-half
        v2f afrag = *(const v2f*)(arow + k0);
#pragma unroll
        for (int n = 0; n < 8; ++n) {
            v2f bfrag = *(const v2f*)(&xt[(16 * n + lo) * LDSROW + k0]);
            acc[n] = __builtin_amdgcn_wmma_f32_16x16x4_f32(
                false, afrag, false, bfrag, (short)0, acc[n], false, false);
        }
    }

#pragma unroll
    for (int n = 0; n < 8; ++n) {
#pragma unroll
        for (int r = 0; r < 8; ++r) {
            const int q = wave * 16 + r + 8 * hi;
            const int s = s0 + 16 * n + lo;
            eb[(size_t)q * S_ + s] = acc[n][r];
        }
    }
}

// ---------------------------------------------------------------------------
// Kernel 2: per-(b,q) online softmax stats over S: stats = {rowmax, 1/sumexp}
// ---------------------------------------------------------------------------
__global__ __launch_bounds__(256)
void row_stats(const float* __restrict__ energy, float* __restrict__ stats) {
    const int row = blockIdx.x;                    // b*Q_ + q
    const float* p = energy + (size_t)row * S_;
    float m = -INFINITY, s = 0.f;
    for (int i = threadIdx.x; i < S_; i += 256) {
        const float v = p[i];
        const float nm = fmaxf(m, v);
        s = s * __expf(m - nm) + __expf(v - nm);
        m = nm;
    }
    __shared__ float sm[256], ss[256];
    sm[threadIdx.x] = m; ss[threadIdx.x] = s;
    __syncthreads();
    for (int off = 128; off; off >>= 1) {
        if (threadIdx.x < off) {
            const float m1 = sm[threadIdx.x],       s1 = ss[threadIdx.x];
            const float m2 = sm[threadIdx.x + off], s2 = ss[threadIdx.x + off];
            const float nm = fmaxf(m1, m2);
            sm[threadIdx.x] = nm;
            ss[threadIdx.x] = s1 * __expf(m1 - nm) + s2 * __expf(m2 - nm);
        }
        __syncthreads();
    }
    if (threadIdx.x == 0) {
        stats[2 * row]     = sm[0];
        stats[2 * row + 1] = 1.0f / ss[0];
    }
}

// ---------------------------------------------------------------------------
// Kernel 3: summary[b,q,e] += sum_s softmax(energy)[b,q,s] * x[b,e,s]
// x tile -> LDS via TDM (row stride S_, padded rows); P tile = exp-normalized
// energy staged by VALU. M=Q, N=E, K=s; fragments are contiguous b64 loads.
// ---------------------------------------------------------------------------
__global__ __launch_bounds__(256)
void gemm2_summary(const float* __restrict__ x, const float* __restrict__ energy,
                   const float* __restrict__ stats, float* __restrict__ summary) {
    extern __shared__ float lds[];
    float* xl = lds;                     // [E_][LDSROW]  x: [e][s_local]
    float* pl = lds + E_ * LDSROW;       // [Q_][LDSROW]  P: [q][s_local]
    const int blk  = blockIdx.x;
    const int b    = blk / (S_ / SCHUNK);
    const int sbeg = (blk % (S_ / SCHUNK)) * SCHUNK;
    const float* xb  = x      + (size_t)b * E_ * S_;
    const float* eb  = energy + (size_t)b * Q_ * S_;
    const float* stb = stats  + (size_t)b * Q_ * 2;
    float*       ob  = summary + (size_t)b * Q_ * E_;

    const int tid = threadIdx.x;
    const int wave = tid >> 5, lane = tid & 31;
    const int hi = lane >> 4, lo = lane & 15;

    v8f acc[8] = {};
    for (int s0 = sbeg; s0 < sbeg + SCHUNK; s0 += STILE) {
        __syncthreads();                 // protect LDS reuse from previous iter
#ifdef HAVE_TDM
        if (tid < 32) {
            tdm_load_2d(xb + s0, (unsigned)(uintptr_t)xl, E_, STILE, S_);
        }
#else
        for (int idx = tid; idx < E_ * (STILE / 4); idx += 256) {
            const int e  = idx >> 5;
            const int c4 = (idx & 31) << 2;
            *(float4*)(&xl[e * LDSROW + c4]) =
                *(const float4*)(xb + (size_t)e * S_ + s0 + c4);
        }
#endif
        // stage P tile = exp(energy - max) * invZ (coalesced read + VALU exp)
        for (int idx = tid; idx < Q_ * (STILE / 4); idx += 256) {
            const int q  = idx >> 5;
            const int c4 = (idx & 31) << 2;
            float4 v = *(const float4*)(eb + (size_t)q * S_ + s0 + c4);
            const float mx = stb[2 * q], iz = stb[2 * q + 1];
            v.x = __expf(v.x - mx) * iz;
            v.y = __expf(v.y - mx) * iz;
            v.z = __expf(v.z - mx) * iz;
            v.w = __expf(v.w - mx) * iz;
            *(float4*)(&pl[q * LDSROW + c4]) = v;
        }
#ifdef HAVE_TDM
        if (tid < 32) __builtin_amdgcn_s_wait_tensorcnt(0);
#endif
        __syncthreads();

        const float* arow = &pl[(16 * wave + lo) * LDSROW];
#pragma unroll 4
        for (int stp = 0; stp < STILE / 4; ++stp) {
            const int k0 = 4 * stp + 2 * hi;             // s_local pair
            v2f afrag = *(const v2f*)(arow + k0);
#pragma unroll
            for (int n = 0; n < 8; ++n) {
                v2f bfrag = *(const v2f*)(&xl[(16 * n + lo) * LDSROW + k0]);
                acc[n] = __builtin_amdgcn_wmma_f32_16x16x4_f32(
                    false, afrag, false, bfrag, (short)0, acc[n], false, false);
            }
        }
    }

#pragma unroll
    for (int n = 0; n < 8; ++n) {
#pragma unroll
        for (int r = 0; r < 8; ++r) {
            const int q = 16 * wave + r + 8 * hi;
            const int e = 16 * n + lo;
            atomicAdd(&ob[(size_t)q * E_ + e], acc[n][r]);
        }
    }
}

// ---------------------------------------------------------------------------
extern "C" void kernel_launch(void* const* d_in, const int* in_sizes, int n_in,
                              void* d_out, int out_size, void* d_ws, size_t ws_size,
                              hipStream_t stream) {
    (void)in_sizes; (void)n_in; (void)out_size; (void)ws_size;
    const float* x  = (const float*)d_in[0];
    const float* Kq = (const float*)d_in[1];
    float* energy  = (float*)d_out;                          // B*Q*H*W floats
    float* summary = energy + (size_t)B_ * Q_ * S_;          // B*Q*E floats
    float* stats   = (float*)d_ws;                           // B*Q*2 floats

    (void)hipMemsetAsync(summary, 0, (size_t)B_ * Q_ * E_ * sizeof(float), stream);

    gemm1_energy<<<B_ * (S_ / STILE), 256,
                   (STILE + Q_) * LDSROW * sizeof(float), stream>>>(x, Kq, energy);
    row_stats<<<B_ * Q_, 256, 0, stream>>>(energy, stats);
    gemm2_summary<<<B_ * (S_ / SCHUNK), 256,
                    (E_ + Q_) * LDSROW * sizeof(float), stream>>>(
        x, energy, stats, summary);
}